// RoIAlign_3882650436490
// MI455X (gfx1250) — compile-verified
//
#include <hip/hip_runtime.h>

// RoIAlign (legacy 7x7, (AH-1) grid, valid-masked) for MI455X / gfx1250.
// Memory-bound gather kernel: per-(roi,bin) metadata broadcast through LDS,
// fully coalesced b32 stores, per-lane bilinear gather out of L2-resident
// feature map. CDNA5 paths: wave32, async global->LDS stage of the RoI row
// (+ s_wait_asynccnt), ds_* metadata broadcast, NT-hinted output stores so
// the 100 MB output stream does not evict the 156 MiB feature set from the
// 192 MB L2.

constexpr int   C_    = 256;
constexpr int   H_    = 200;
constexpr int   W_    = 200;
constexpr int   AH_   = 7;
constexpr int   AW_   = 7;
constexpr int   NBINS = AH_ * AW_;     // 49
constexpr int   HW_   = H_ * W_;       // 40000
constexpr int   CHW_  = C_ * HW_;      // 10,240,000
constexpr float SCALE_ = 0.25f;

constexpr int SPLIT = 4;               // channel slices per RoI -> 8000 blocks
constexpr int CPB   = C_ / SPLIT;      // 64 channels per block
constexpr int BLOCK = 256;             // 8 wave32 per block

#if defined(__HIP_DEVICE_COMPILE__) && __has_builtin(__builtin_amdgcn_global_load_async_to_lds_b32)
#define HAVE_ASYNC_LDS 1
typedef __attribute__((address_space(1))) int* as1_i32p;
typedef __attribute__((address_space(3))) int* as3_i32p;
#endif

__global__ __launch_bounds__(BLOCK)
void roialign_gfx1250_kernel(const float* __restrict__ feat,
                             const float* __restrict__ rois,
                             float* __restrict__ out)
{
    __shared__ float s_roi[8];
    __shared__ int   s_base[NBINS];
    __shared__ float s_w00[NBINS];
    __shared__ float s_w01[NBINS];
    __shared__ float s_w10[NBINS];
    __shared__ float s_w11[NBINS];

    const int r   = blockIdx.x;
    const int c0  = blockIdx.y * CPB;
    const int tid = threadIdx.x;

    // ---- Phase 0: stage this RoI's 5 floats into LDS (CDNA5 async copy) ----
#if defined(HAVE_ASYNC_LDS)
    if (tid < 5) {
        __builtin_amdgcn_global_load_async_to_lds_b32(
            (as1_i32p)(void*)(const_cast<float*>(rois) + 5ull * (unsigned)r + tid),
            (as3_i32p)(void*)(&s_roi[tid]),
            /*offset=*/0, /*cpol=*/0);
    }
#if __has_builtin(__builtin_amdgcn_s_wait_asynccnt)
    __builtin_amdgcn_s_wait_asynccnt(0);
#else
    asm volatile("s_wait_asynccnt 0x0" ::: "memory");
#endif
#else
    if (tid < 5) s_roi[tid] = rois[5ull * (unsigned)r + tid];
#endif
    __syncthreads();

    // ---- Phase 1: 49 threads compute per-bin base offset + bilinear weights ----
    if (tid < NBINS) {
        const float bf = s_roi[0];
        const float x1 = s_roi[1] * SCALE_;
        const float y1 = s_roi[2] * SCALE_;
        const float x2 = s_roi[3] * SCALE_;
        const float y2 = s_roi[4] * SCALE_;

        const float roi_w = fmaxf(x2 - x1 + 1.0f, 0.0f);
        const float roi_h = fmaxf(y2 - y1 + 1.0f, 0.0f);
        const float bin_h = roi_h / (float)(AH_ - 1);
        const float bin_w = roi_w / (float)(AW_ - 1);

        const int ph = tid / AW_;
        const int pw = tid - ph * AW_;

        const float h = y1 + (float)ph * bin_h;
        const float w = x1 + (float)pw * bin_w;

        const bool valid = (h >= 0.0f) && (h < (float)H_) &&
                           (w >= 0.0f) && (w < (float)W_);

        const float h0f = fmaxf(fminf(floorf(h), (float)(H_ - 2)), 0.0f);
        const float w0f = fmaxf(fminf(floorf(w), (float)(W_ - 2)), 0.0f);
        const int   h0  = (int)h0f;
        const int   w0  = (int)w0f;
        const float hr  = h - h0f;
        const float wr  = w - w0f;
        const float vv  = valid ? 1.0f : 0.0f;

        s_base[tid] = (int)bf * CHW_ + h0 * W_ + w0;
        s_w00[tid]  = (1.0f - hr) * (1.0f - wr) * vv;
        s_w01[tid]  = (1.0f - hr) * wr          * vv;
        s_w10[tid]  = hr          * (1.0f - wr) * vv;
        s_w11[tid]  = hr          * wr          * vv;
    }
    __syncthreads();

    // ---- Phase 2: coalesced gather/interp/store over 64 channels x 49 bins ----
    // Output layout (R, C, 7, 7): linear index within RoI is c*49 + bin, so
    // consecutive threads -> consecutive output addresses (coalesced b32 stores).
    // Output is write-once: store non-temporal so it doesn't evict features
    // from L2 (features ~156 MiB, output ~100 MB, L2 = 192 MB).
    const long long out_base = (long long)r * (C_ * NBINS) + (long long)c0 * NBINS;

    for (int i = tid; i < CPB * NBINS; i += BLOCK) {
        const int c   = i / NBINS;
        const int bin = i - c * NBINS;

        const int   base = s_base[bin] + (c0 + c) * HW_;
        const float w00  = s_w00[bin];
        const float w01  = s_w01[bin];
        const float w10  = s_w10[bin];
        const float w11  = s_w11[bin];

        const float* __restrict__ p = feat + base;
        const float v00 = p[0];
        const float v01 = p[1];
        const float v10 = p[W_];
        const float v11 = p[W_ + 1];

        const float res =
            fmaf(v00, w00, fmaf(v01, w01, fmaf(v10, w10, v11 * w11)));
        __builtin_nontemporal_store(res, out + out_base + i);
    }
}

extern "C" void kernel_launch(void* const* d_in, const int* in_sizes, int n_in,
                              void* d_out, int out_size, void* d_ws, size_t ws_size,
                              hipStream_t stream)
{
    const float* feat = (const float*)d_in[0];
    const float* rois = (const float*)d_in[1];
    float*       out  = (float*)d_out;

    const int R = in_sizes[1] / 5;   // 2000

    dim3 grid((unsigned)R, SPLIT, 1);
    dim3 block(BLOCK, 1, 1);
    hipLaunchKernelGGL(roialign_gfx1250_kernel, grid, block, 0, stream,
                       feat, rois, out);
}